// CellGate_77567109366012
// MI455X (gfx1250) — compile-verified
//
#include <hip/hip_runtime.h>
#include <math.h>

typedef __attribute__((ext_vector_type(2))) float v2f;
typedef __attribute__((ext_vector_type(8))) float v8f;

#define DCH 64
#define LDW 68              // LDS row stride in floats (bank-conflict-free, 16B aligned)
#define TPW 8               // 16-row tiles per block (one per wave)
#define TPB 256

// ---------------------------------------------------------------------------
// helpers: stage weights / tiles into LDS
// ---------------------------------------------------------------------------
__device__ __forceinline__ void stage_w(const float* __restrict__ W, float* sW) {
  for (int i = threadIdx.x; i < DCH * 16; i += TPB) {   // 64 rows x 16 float4
    int r = i >> 4, c = (i & 15) << 2;
    float4 v = *(const float4*)(W + r * DCH + c);
    *(float4*)(sW + r * LDW + c) = v;
  }
}

__device__ __forceinline__ void stage_w_sum(const float* __restrict__ A,
                                            const float* __restrict__ B, float* sW) {
  for (int i = threadIdx.x; i < DCH * 16; i += TPB) {
    int r = i >> 4, c = (i & 15) << 2;
    float4 a = *(const float4*)(A + r * DCH + c);
    float4 b = *(const float4*)(B + r * DCH + c);
    float4 v = make_float4(a.x + b.x, a.y + b.y, a.z + b.z, a.w + b.w);
    *(float4*)(sW + r * LDW + c) = v;
  }
}

__device__ __forceinline__ void stage_x(const float* __restrict__ X, int row0,
                                        float* sX, int lane) {
  for (int i = lane; i < 16 * 16; i += 32) {            // 16 rows x 16 float4
    int r = i >> 4, c = (i & 15) << 2;
    float4 v = *(const float4*)(X + (size_t)(row0 + r) * DCH + c);
    *(float4*)(sX + r * LDW + c) = v;
  }
}

__device__ __forceinline__ void stage_mean(const float* __restrict__ acc,
                                           const float* __restrict__ cnt, int row0,
                                           float* sX, int lane) {
  for (int i = lane; i < 16 * 16; i += 32) {
    int r = i >> 4, c = (i & 15) << 2;
    float inv = 1.0f / fmaxf(cnt[row0 + r], 1.0f);
    float4 v = *(const float4*)(acc + (size_t)(row0 + r) * DCH + c);
    v.x *= inv; v.y *= inv; v.z *= inv; v.w *= inv;
    *(float4*)(sX + r * LDW + c) = v;
  }
}

// ---------------------------------------------------------------------------
// one 16x64 = [16x64] @ [64x64]^T term, accumulated into 4x v8f via
// v_wmma_f32_16x16x4_f32 (K-loop of 16, 4 column tiles)
// A layout: lane(0-15)=M, VGPRr: K = 4*ks + 2*(lane/16) + r  (X row per lane)
// B layout: lane%16 = N (out col), same K striping         (W row per lane)
// ---------------------------------------------------------------------------
__device__ __forceinline__ void wmma_term(const float* sX, const float* sW,
                                          v8f d[4], int lane) {
  const int lrow = lane & 15;
  const int kh   = (lane >> 4) << 1;       // 0 or 2
  const float* xrow = sX + lrow * LDW;
#pragma unroll
  for (int nt = 0; nt < 4; ++nt) {
    const float* wrow = sW + (nt * 16 + lrow) * LDW;
    v8f c = d[nt];
#pragma unroll
    for (int ks = 0; ks < 16; ++ks) {
      int k = ks * 4 + kh;
      v2f a; a.x = xrow[k]; a.y = xrow[k + 1];
      v2f b; b.x = wrow[k]; b.y = wrow[k + 1];
      c = __builtin_amdgcn_wmma_f32_16x16x4_f32(false, a, false, b, (short)0, c,
                                                false, false);
    }
    d[nt] = c;
  }
}

// D layout: lane%16 = N, VGPR v -> M = 8*(lane/16) + v
__device__ __forceinline__ void store_tile(float* __restrict__ out, int row0,
                                           const v8f d[4], const float* sB, int lane) {
  int n  = lane & 15;
  int mb = (lane >> 4) << 3;
#pragma unroll
  for (int nt = 0; nt < 4; ++nt) {
    float bias = sB ? sB[nt * 16 + n] : 0.0f;
#pragma unroll
    for (int v = 0; v < 8; ++v)
      out[(size_t)(row0 + mb + v) * DCH + nt * 16 + n] = d[nt][v] + bias;
  }
}

// ---------------------------------------------------------------------------
// kernels
// ---------------------------------------------------------------------------
__global__ void zero_f4(float4* p, int n4) {
  int i = blockIdx.x * blockDim.x + threadIdx.x;
  if (i < n4) p[i] = make_float4(0.f, 0.f, 0.f, 0.f);
}

__global__ void count_deg(const int* __restrict__ dst, float* cnt, int E) {
  int e = blockIdx.x * blockDim.x + threadIdx.x;
  if (e < E) atomicAdd(&cnt[dst[e]], 1.0f);
}

// 64 lanes per edge: coalesced 256B gather + 64 coalesced f32 atomic adds
__global__ void scatter_add(const float* __restrict__ tsrc, const int* __restrict__ src,
                            const int* __restrict__ dst, float* acc, int E) {
  long gid = (long)blockIdx.x * blockDim.x + threadIdx.x;
  int e = (int)(gid >> 6);
  int c = (int)(gid & 63);
  if (e >= E) return;
  int s = src[e];
  int d = dst[e];
  atomicAdd(&acc[(size_t)d * DCH + c], tsrc[(size_t)s * DCH + c]);
}

// t = x @ W^T
__global__ void __launch_bounds__(TPB) proj_kernel(const float* __restrict__ x,
                                                   const float* __restrict__ W,
                                                   float* __restrict__ t, int nrows) {
  __shared__ float sW[DCH * LDW];
  __shared__ float sX[TPW][16 * LDW];
  int lane = threadIdx.x & 31, wave = threadIdx.x >> 5;
  int row0 = (blockIdx.x * TPW + wave) * 16;
  bool active = row0 < nrows;
  stage_w(W, sW);
  if (active) stage_x(x, row0, sX[wave], lane);
  __syncthreads();
  v8f d[4] = {};
  if (active) {
    wmma_term(sX[wave], sW, d, lane);
    store_tile(t, row0, d, nullptr, lane);
  }
}

// t = mean(acc,cnt) @ Wl^T + bl + t @ Wr^T    (in place on t, row-local)
__global__ void __launch_bounds__(TPB) combine1_kernel(
    const float* __restrict__ acc, const float* __restrict__ cnt,
    const float* __restrict__ Wl, const float* __restrict__ bl,
    const float* __restrict__ Wr, float* __restrict__ t, int nrows) {
  __shared__ float sW[DCH * LDW];
  __shared__ float sX[TPW][16 * LDW];
  __shared__ float sB[DCH];
  int lane = threadIdx.x & 31, wave = threadIdx.x >> 5;
  int row0 = (blockIdx.x * TPW + wave) * 16;
  bool active = row0 < nrows;
  if (threadIdx.x < DCH) sB[threadIdx.x] = bl[threadIdx.x];
  v8f d[4] = {};
  stage_w(Wl, sW);
  if (active) stage_mean(acc, cnt, row0, sX[wave], lane);
  __syncthreads();
  if (active) wmma_term(sX[wave], sW, d, lane);
  __syncthreads();
  stage_w(Wr, sW);
  if (active) stage_x(t, row0, sX[wave], lane);
  __syncthreads();
  if (active) {
    wmma_term(sX[wave], sW, d, lane);
    store_tile(t, row0, d, sB, lane);
  }
}

// t = mean1 @ Wl1^T + mean2 @ Wl2^T + t @ (Wr1+Wr2)^T + (bl1+bl2)
__global__ void __launch_bounds__(TPB) combine2_kernel(
    const float* __restrict__ acc1, const float* __restrict__ cnt1,
    const float* __restrict__ Wl1, const float* __restrict__ bl1,
    const float* __restrict__ acc2, const float* __restrict__ cnt2,
    const float* __restrict__ Wl2, const float* __restrict__ bl2,
    const float* __restrict__ Wr1, const float* __restrict__ Wr2,
    float* __restrict__ t, int nrows) {
  __shared__ float sW[DCH * LDW];
  __shared__ float sX[TPW][16 * LDW];
  __shared__ float sB[DCH];
  int lane = threadIdx.x & 31, wave = threadIdx.x >> 5;
  int row0 = (blockIdx.x * TPW + wave) * 16;
  bool active = row0 < nrows;
  if (threadIdx.x < DCH) sB[threadIdx.x] = bl1[threadIdx.x] + bl2[threadIdx.x];
  v8f d[4] = {};
  stage_w(Wl1, sW);
  if (active) stage_mean(acc1, cnt1, row0, sX[wave], lane);
  __syncthreads();
  if (active) wmma_term(sX[wave], sW, d, lane);
  __syncthreads();
  stage_w(Wl2, sW);
  if (active) stage_mean(acc2, cnt2, row0, sX[wave], lane);
  __syncthreads();
  if (active) wmma_term(sX[wave], sW, d, lane);
  __syncthreads();
  stage_w_sum(Wr1, Wr2, sW);
  if (active) stage_x(t, row0, sX[wave], lane);
  __syncthreads();
  if (active) {
    wmma_term(sX[wave], sW, d, lane);
    store_tile(t, row0, d, sB, lane);
  }
}

// out = f*c + i*tanh(bias[ch] + t)
__global__ void gate_kernel(const float* __restrict__ t, const float* __restrict__ bias,
                            const float* __restrict__ f, const float* __restrict__ cc,
                            const float* __restrict__ ii, float* __restrict__ out, int n) {
  int idx = blockIdx.x * blockDim.x + threadIdx.x;
  if (idx < n) {
    int ch = idx & (DCH - 1);
    out[idx] = f[idx] * cc[idx] + ii[idx] * tanhf(bias[ch] + t[idx]);
  }
}

// ---------------------------------------------------------------------------
extern "C" void kernel_launch(void* const* d_in, const int* in_sizes, int n_in,
                              void* d_out, int out_size, void* d_ws, size_t ws_size,
                              hipStream_t stream) {
  const float* x_A    = (const float*)d_in[0];
  const float* x_B    = (const float*)d_in[1];
  const float* c_A    = (const float*)d_in[4];
  const float* c_B    = (const float*)d_in[5];
  const float* i_A    = (const float*)d_in[6];
  const float* i_B    = (const float*)d_in[7];
  const float* f_A    = (const float*)d_in[8];
  const float* f_B    = (const float*)d_in[9];
  const float* Wx_A   = (const float*)d_in[10];
  const float* Wx_B   = (const float*)d_in[11];
  const float* bias_A = (const float*)d_in[12];
  const float* bias_B = (const float*)d_in[13];
  const float* Wl     = (const float*)d_in[14];  // [L,3,64,64]
  const float* Wr     = (const float*)d_in[15];
  const float* bl     = (const float*)d_in[16];  // [L,3,64]
  const int* eAB = (const int*)d_in[17];
  const int* eBA = (const int*)d_in[18];
  const int* eAA = (const int*)d_in[19];

  const int N  = in_sizes[0] / DCH;       // 100000 (multiple of 16)
  const int E  = in_sizes[17] / 2;        // 1600000
  const size_t ND = (size_t)N * DCH;

  float* ws    = (float*)d_ws;
  float* t_A   = ws;
  float* t_B   = ws + ND;
  float* acc0  = ws + 2 * ND;
  float* acc1  = ws + 3 * ND;
  float* cntB  = ws + 4 * ND;             // deg of B via AB
  float* cntA  = cntB + N;                // deg of A via BA
  float* cntAA = cntA + N;                // deg of A via AA

  const int tileBlocks = (N / 16 + TPW - 1) / TPW;
  const int cntBlocks  = (E + 255) / 256;
  const int scatBlocks = (int)(((long)E * 64 + 255) / 256);
  const int ewBlocks   = (int)((ND + 255) / 256);

  // degree counts (layer-invariant)
  zero_f4<<<(3 * N / 4 + 255) / 256, 256, 0, stream>>>((float4*)cntB, 3 * N / 4);
  count_deg<<<cntBlocks, 256, 0, stream>>>(eAB + E, cntB, E);
  count_deg<<<cntBlocks, 256, 0, stream>>>(eBA + E, cntA, E);
  count_deg<<<cntBlocks, 256, 0, stream>>>(eAA + E, cntAA, E);

  // input projections (WMMA)
  proj_kernel<<<tileBlocks, TPB, 0, stream>>>(x_A, Wx_A, t_A, N);
  proj_kernel<<<tileBlocks, TPB, 0, stream>>>(x_B, Wx_B, t_B, N);

  for (int l = 0; l < 2; ++l) {
    const float* Wl0 = Wl + (size_t)(l * 3 + 0) * DCH * DCH;
    const float* Wl1 = Wl + (size_t)(l * 3 + 1) * DCH * DCH;
    const float* Wl2 = Wl + (size_t)(l * 3 + 2) * DCH * DCH;
    const float* Wr0 = Wr + (size_t)(l * 3 + 0) * DCH * DCH;
    const float* Wr1 = Wr + (size_t)(l * 3 + 1) * DCH * DCH;
    const float* Wr2 = Wr + (size_t)(l * 3 + 2) * DCH * DCH;
    const float* bl0 = bl + (size_t)(l * 3 + 0) * DCH;
    const float* bl1 = bl + (size_t)(l * 3 + 1) * DCH;
    const float* bl2 = bl + (size_t)(l * 3 + 2) * DCH;

    // zero both accumulators (contiguous)
    zero_f4<<<(int)((2 * ND / 4 + 255) / 256), 256, 0, stream>>>((float4*)acc0,
                                                                 (int)(2 * ND / 4));
    scatter_add<<<scatBlocks, 256, 0, stream>>>(t_A, eAB, eAB + E, acc0, E); // A->B
    scatter_add<<<scatBlocks, 256, 0, stream>>>(t_B, eBA, eBA + E, acc1, E); // B->A
    // B update (reads old t_B after BA scatter consumed it)
    combine1_kernel<<<tileBlocks, TPB, 0, stream>>>(acc0, cntB, Wl0, bl0, Wr0, t_B, N);
    // reuse acc0 for AA
    zero_f4<<<(int)((ND / 4 + 255) / 256), 256, 0, stream>>>((float4*)acc0,
                                                             (int)(ND / 4));
    scatter_add<<<scatBlocks, 256, 0, stream>>>(t_A, eAA, eAA + E, acc0, E); // A->A
    combine2_kernel<<<tileBlocks, TPB, 0, stream>>>(acc1, cntA, Wl1, bl1,
                                                    acc0, cntAA, Wl2, bl2,
                                                    Wr1, Wr2, t_A, N);
  }

  float* out_A = (float*)d_out;
  float* out_B = out_A + ND;
  gate_kernel<<<ewBlocks, 256, 0, stream>>>(t_A, bias_A, f_A, c_A, i_A, out_A, (int)ND);
  gate_kernel<<<ewBlocks, 256, 0, stream>>>(t_B, bias_B, f_B, c_B, i_B, out_B, (int)ND);
}